// EmoShift_31284541784731
// MI455X (gfx1250) — compile-verified
//
#include <hip/hip_runtime.h>

typedef float v2f __attribute__((ext_vector_type(2)));
typedef float v8f __attribute__((ext_vector_type(8)));

#define SS   96
#define BB   16
#define DD   512
#define HID  128
#define KCH  64

// ---------------------------------------------------------------------------
// Stage 1:  HA[1536,128] = E[1536,512] * W1a^T,   HB = Ecmp * W1b^T
// grid (96, 1, 2), block 256 (8 waves).  Each wave: one 16x16 C tile via
// v_wmma_f32_16x16x4_f32, K streamed through LDS in chunks of 64.
// ---------------------------------------------------------------------------
__global__ __launch_bounds__(256) void emoshift_gemm_stage1(
    const float* __restrict__ E,        // (96,16,512) flat = (1536,512)
    const float* __restrict__ Ecmp,     // (96,16,512)
    const float* __restrict__ W1,       // (128, 1024)
    float* __restrict__ HA,             // (1536,128)
    float* __restrict__ HB)             // (1536,128)
{
    __shared__ float ldsA[KCH][17];     // [k][row], pad -> conflict-free reads
    __shared__ float ldsB[KCH][129];    // [k][n],   pad -> conflict-free reads

    const int z      = blockIdx.z;
    const float* Es  = z ? Ecmp : E;
    float*       Hd  = z ? HB   : HA;
    const int dOff   = z * DD;          // W1a = W1[:, :512], W1b = W1[:, 512:]

    const int tid   = threadIdx.x;
    const int lane  = tid & 31;
    const int wave  = tid >> 5;
    const int mBase = blockIdx.x * 16;
    const int nBase = wave * 16;

    v8f acc = {0.f, 0.f, 0.f, 0.f, 0.f, 0.f, 0.f, 0.f};

    const int half = lane >> 4;         // lane half selects K pair (+0 / +2)
    const int rn   = lane & 15;         // A row / B column within tile

    for (int kc = 0; kc < DD; kc += KCH) {
        // stage A chunk: rows mBase..+15, cols kc..+63  (coalesced global reads)
        for (int idx = tid; idx < 16 * KCH; idx += 256) {
            int row = idx >> 6;
            int k   = idx & 63;
            ldsA[k][row] = Es[(mBase + row) * DD + kc + k];
        }
        // stage B chunk: W1 rows n=0..127, cols dOff+kc..+63
        for (int idx = tid; idx < HID * KCH; idx += 256) {
            int n = idx >> 6;
            int k = idx & 63;
            ldsB[k][n] = W1[n * (2 * DD) + dOff + kc + k];
        }
        __syncthreads();

        #pragma unroll
        for (int k0 = 0; k0 < KCH; k0 += 4) {
            const int ka = k0 + (half << 1);
            v2f a, b;
            // A 16x4 layout (ISA 7.12.2): VGPR0=K{0,2}, VGPR1=K{1,3} by lane half
            a.x = ldsA[ka + 0][rn];
            a.y = ldsA[ka + 1][rn];
            // B 4x16: rows striped across lanes; VGPR/lane-half select K, mirrored
            b.x = ldsB[ka + 0][nBase + rn];
            b.y = ldsB[ka + 1][nBase + rn];
            acc = __builtin_amdgcn_wmma_f32_16x16x4_f32(
                false, a, false, b, (short)0, acc, false, false);
        }
        __syncthreads();
    }

    // C 16x16 f32 layout: VGPR v -> M=v (lanes 0-15), M=v+8 (lanes 16-31)
    #pragma unroll
    for (int v = 0; v < 8; ++v) {
        const int m = mBase + v + half * 8;
        Hd[m * HID + nBase + rn] = acc[v];
    }
}

// ---------------------------------------------------------------------------
// Stage 2: logits[i,j,b,c] = W2[c,:] . relu(ha[i,b,:] + hb[j,b,:] + b1) + b2[c]
// One wave32 per (i,j,b); each lane owns 4 hidden units (float4, coalesced).
// ---------------------------------------------------------------------------
__global__ __launch_bounds__(256) void emoshift_pair_head(
    const float* __restrict__ HA, const float* __restrict__ HB,
    const float* __restrict__ b1, const float* __restrict__ W2,
    const float* __restrict__ b2, float* __restrict__ out)
{
    const int gw   = blockIdx.x * 8 + (threadIdx.x >> 5);   // wave id
    const int lane = threadIdx.x & 31;
    const int b = gw & (BB - 1);
    const int j = (gw >> 4) % SS;
    const int i = gw / (BB * SS);

    const float4 ha = *(const float4*)(HA + (i * BB + b) * HID + lane * 4);
    const float4 hb = *(const float4*)(HB + (j * BB + b) * HID + lane * 4);
    const float4 bv = *(const float4*)(b1 + lane * 4);
    const float4 w0 = *(const float4*)(W2 + lane * 4);
    const float4 w1 = *(const float4*)(W2 + HID + lane * 4);

    const float h0 = fmaxf(ha.x + hb.x + bv.x, 0.f);
    const float h1 = fmaxf(ha.y + hb.y + bv.y, 0.f);
    const float h2 = fmaxf(ha.z + hb.z + bv.z, 0.f);
    const float h3 = fmaxf(ha.w + hb.w + bv.w, 0.f);

    float a0 = h0 * w0.x + h1 * w0.y + h2 * w0.z + h3 * w0.w;
    float a1 = h0 * w1.x + h1 * w1.y + h2 * w1.z + h3 * w1.w;

    #pragma unroll
    for (int off = 16; off > 0; off >>= 1) {
        a0 += __shfl_xor(a0, off, 32);
        a1 += __shfl_xor(a1, off, 32);
    }

    if (lane == 0) {
        float* o = out + ((i * SS + j) * BB + b) * 2;
        o[0] = a0 + b2[0];
        o[1] = a1 + b2[1];
    }
}

// ---------------------------------------------------------------------------
extern "C" void kernel_launch(void* const* d_in, const int* in_sizes, int n_in,
                              void* d_out, int out_size, void* d_ws, size_t ws_size,
                              hipStream_t stream)
{
    (void)in_sizes; (void)n_in; (void)out_size; (void)ws_size;
    const float* embeds     = (const float*)d_in[0];
    // d_in[1] umask, d_in[2] qmask: unused by the reference computation
    const float* embeds_cmp = (const float*)d_in[3];
    const float* W1         = (const float*)d_in[4];
    const float* b1         = (const float*)d_in[5];
    const float* W2         = (const float*)d_in[6];
    const float* b2         = (const float*)d_in[7];

    float* HA = (float*)d_ws;                       // 1536*128 floats
    float* HB = HA + (size_t)SS * BB * HID;         // 1536*128 floats

    dim3 g1(SS * BB / 16, 1, 2);                    // 96 M-tiles, z: ha vs hb
    emoshift_gemm_stage1<<<g1, dim3(256), 0, stream>>>(embeds, embeds_cmp, W1, HA, HB);

    dim3 g2(SS * SS * BB / 8);                      // 147456 waves / 8 per block
    emoshift_pair_head<<<g2, dim3(256), 0, stream>>>(HA, HB, b1, W2, b2, (float*)d_out);
}